// NeighborPointsGenerator_63419487092879
// MI455X (gfx1250) — compile-verified
//
#include <hip/hip_runtime.h>
#include <math.h>

#define BS 4
#define H 256
#define W 256
#define HW (H * W)
#define G 500
#define GP 512          // GT count padded to 32 tiles of 16
#define NT (GP / 16)    // 32 gt tiles per pixel tile
#define K_RING 120
#define DIS_THR2 400.0f // (STRIDE*RADIUS)^2

typedef __attribute__((ext_vector_type(2))) float v2f;
typedef __attribute__((ext_vector_type(8))) float v8f;

// NaN-free fast mins (inputs are finite by construction: coords in [0,255],
// GT coords in [0,255], pad sentinel 1e30) — avoids fminf's canonicalize ops.
__device__ __forceinline__ float min_num(float a, float b) {
  float r;
  asm("v_min_num_f32 %0, %1, %2" : "=v"(r) : "v"(a), "v"(b));
  return r;
}
__device__ __forceinline__ float min3_num(float a, float b, float c) {
  float r;
  asm("v_min3_num_f32 %0, %1, %2, %3" : "=v"(r) : "v"(a), "v"(b), "v"(c));
  return r;
}

// ---------------------------------------------------------------------------
// Kernel 1: build padded B panel: B[:,n] = (-2gx, -2gy, 1, gx^2+gy^2) for
// valid slots, (0,0,0,1e30) for invalid/padded (ragged mask baked into GEMM).
// Also compute the 120 ring offsets in double precision (matches numpy f64).
// ---------------------------------------------------------------------------
__global__ void npg_prep_kernel(const float* __restrict__ gt,
                                const int* __restrict__ gt_nums,
                                float* __restrict__ Bp,
                                int* __restrict__ offs) {
  int tid = blockIdx.x * blockDim.x + threadIdx.x;
  if (tid < BS * GP) {
    int b = tid / GP, s = tid % GP;
    float o0 = 0.f, o1 = 0.f, o2 = 0.f, o3 = 1.0e30f;
    if (s < G && s < gt_nums[b]) {
      float gx = gt[(b * G + s) * 2 + 0];
      float gy = gt[(b * G + s) * 2 + 1];
      o0 = -2.0f * gx;
      o1 = -2.0f * gy;
      o2 = 1.0f;
      o3 = gx * gx + gy * gy;
    }
    float4 v = make_float4(o0, o1, o2, o3);
    *(float4*)(Bp + (size_t)tid * 4) = v;
  }
  if (tid < K_RING) {
    // ring i has 8*(i+1) points at radius 4*(i+1); cumulative starts:
    const int cum[6] = {0, 8, 24, 48, 80, 120};
    int i = 0;
    while (tid >= cum[i + 1]) ++i;
    int n = 8 * (i + 1);
    int j = tid - cum[i];
    double r = (double)(4 * (i + 1));
    double ang = (double)j / (double)n * 6.283185307179586476925286766559;
    double dx = r * cos(ang);
    double dy = r * sin(ang);
    offs[tid] = (int)rint(dy * (double)W + dx * (double)H);
  }
}

// ---------------------------------------------------------------------------
// Kernel 2: per 16-pixel tile, min squared distance over all GT points via
// V_WMMA_F32_16X16X4_F32:  D[m][n] = A[m]·B[:,n] = p^2 + g^2 - 2 p.g = d^2.
// A[m] = (px, py, p^2, 1) built in registers from the pixel index.
// 4 tiles per iteration; per-row accumulation = 2 v_min3_num per 4 WMMAs.
// ---------------------------------------------------------------------------
__global__ void __launch_bounds__(256)
npg_negflag_kernel(const float* __restrict__ Bp,
                   float* __restrict__ pixels_out,
                   float* __restrict__ flag) {
  const int lane = threadIdx.x & 31;
  const int wave = threadIdx.x >> 5;
  const int gid = blockIdx.x * 8 + wave;  // 0 .. BS*HW/16 - 1
  const int b = gid >> 12;                // / (HW/16 = 4096)
  const int m0 = (gid & 4095) << 4;       // first pixel of tile

  // A operand (ISA layout: lanes 0-15 hold K={0,1}, lanes 16-31 hold K={2,3})
  const int M = lane & 15;
  const int pix = m0 + M;
  const float px = (float)(pix >> 8);
  const float py = (float)(pix & 255);
  v2f a;
  if (lane < 16) { a.x = px;                a.y = py;   }
  else           { a.x = px * px + py * py; a.y = 1.0f; }

  // B operand: point s = tile*16 + (lane%16), k-pair = (lane>=16)?{2,3}:{0,1}
  const float* bbase =
      Bp + ((size_t)b * GP) * 4 + (size_t)(lane & 15) * 4 + ((lane >> 4) << 1);

  v8f mind;
#pragma unroll
  for (int r = 0; r < 8; ++r) mind[r] = 3.0e38f;

#pragma unroll
  for (int t = 0; t < NT; t += 4) {
    v2f bv0 = *(const v2f*)(bbase + (t + 0) * 64);
    v2f bv1 = *(const v2f*)(bbase + (t + 1) * 64);
    v2f bv2 = *(const v2f*)(bbase + (t + 2) * 64);
    v2f bv3 = *(const v2f*)(bbase + (t + 3) * 64);
    v8f c = {};
    v8f d0 = __builtin_amdgcn_wmma_f32_16x16x4_f32(false, a, false, bv0,
                                                   (short)0, c, false, false);
    v8f d1 = __builtin_amdgcn_wmma_f32_16x16x4_f32(false, a, false, bv1,
                                                   (short)0, c, false, false);
    v8f d2 = __builtin_amdgcn_wmma_f32_16x16x4_f32(false, a, false, bv2,
                                                   (short)0, c, false, false);
    v8f d3 = __builtin_amdgcn_wmma_f32_16x16x4_f32(false, a, false, bv3,
                                                   (short)0, c, false, false);
#pragma unroll
    for (int r = 0; r < 8; ++r) {
      float m012 = min3_num(d0[r], d1[r], d2[r]);
      mind[r] = min3_num(mind[r], m012, d3[r]);
    }
  }

  // Min-reduce over the 16 N-lanes of each half (rows M=r and M=r+8)
#pragma unroll
  for (int off = 1; off < 16; off <<= 1) {
#pragma unroll
    for (int r = 0; r < 8; ++r) {
      float o = __shfl_xor(mind[r], off, 32);
      mind[r] = min_num(mind[r], o);
    }
  }

  // pixels_out: lane L writes pixel m0+(L>>1), component L&1
  {
    int p = m0 + (lane >> 1);
    float vx = (float)(p >> 8), vy = (float)(p & 255);
    pixels_out[((size_t)b * HW + p) * 2 + (lane & 1)] = (lane & 1) ? vy : vx;
  }

  // flag: lanes 0 and 16 each own 8 rows of the tile
  if ((lane & 15) == 0) {
    int half = lane >> 4;
#pragma unroll
    for (int r = 0; r < 8; ++r) {
      int row = r + half * 8;
      flag[(size_t)b * HW + m0 + row] = (mind[r] > DIS_THR2) ? -1.0f : 0.0f;
    }
  }
}

// ---------------------------------------------------------------------------
// Kernel 3: positive scatter — flag[b, round(gx*W+gy*H) + ring_off[k]] = 1.0
// for valid g, in-range indices. Races are benign (all write 1.0).
// ---------------------------------------------------------------------------
__global__ void npg_scatter_kernel(const float* __restrict__ gt,
                                   const int* __restrict__ gt_nums,
                                   const int* __restrict__ offs,
                                   float* __restrict__ flag) {
  int tid = blockIdx.x * blockDim.x + threadIdx.x;
  if (tid >= BS * G * K_RING) return;
  int k = tid % K_RING;
  int g = (tid / K_RING) % G;
  int b = tid / (K_RING * G);
  if (g >= gt_nums[b]) return;
  float gx = gt[(b * G + g) * 2 + 0];
  float gy = gt[(b * G + g) * 2 + 1];
  int base = (int)rintf(gx * (float)W + gy * (float)H);
  int idx = base + offs[k];
  if (idx >= 0 && idx < HW) flag[(size_t)b * HW + idx] = 1.0f;
}

// ---------------------------------------------------------------------------
extern "C" void kernel_launch(void* const* d_in, const int* in_sizes, int n_in,
                              void* d_out, int out_size, void* d_ws,
                              size_t ws_size, hipStream_t stream) {
  // inputs: [0]=images (unused, shape only), [1]=gt_points f32, [2]=gt_nums i32
  const float* gt = (const float*)d_in[1];
  const int* gt_nums = (const int*)d_in[2];

  float* out = (float*)d_out;
  float* pixels_out = out;                       // BS*HW*2 floats
  float* flag = out + (size_t)BS * HW * 2;       // BS*HW floats

  float* Bp = (float*)d_ws;                      // BS*GP*4 floats = 32 KB
  int* offs = (int*)((char*)d_ws + (size_t)BS * GP * 4 * sizeof(float));

  npg_prep_kernel<<<(BS * GP + 255) / 256, 256, 0, stream>>>(gt, gt_nums, Bp,
                                                             offs);
  npg_negflag_kernel<<<(BS * HW / 16) / 8, 256, 0, stream>>>(Bp, pixels_out,
                                                             flag);
  npg_scatter_kernel<<<(BS * G * K_RING + 255) / 256, 256, 0, stream>>>(
      gt, gt_nums, offs, flag);
}